// MYPRECONDITIONER_4733053960807
// MI455X (gfx1250) — compile-verified
//
#include <hip/hip_runtime.h>

// GraphNet (3 message-passing layers) for MI455X / gfx1250, wave32.
// MLP layer-1 GEMMs run on the matrix pipe via V_WMMA_F32_16X16X4_F32
// (full FP32: the op is L2-resident & latency-bound, so no reason to downcast).

typedef __attribute__((ext_vector_type(2))) float v2f;
typedef __attribute__((ext_vector_type(8))) float v8f;

#define GN_N_NODES 100000
#define GN_N_EDGES 1600000
#define GN_NODE_F  8
#define GN_EDGE_F  2
#define GN_HID     16
#define GN_EDGE_IN 19   // padded to 20 (5 K-chunks of 4)
#define GN_NODE_IN 11   // padded to 12 (3 K-chunks of 4)

__device__ __forceinline__ v8f gn_wmma_k4(v2f a, v2f b, v8f c) {
  // D(16x16,f32) += A(16x4,f32) * B(4x16,f32)
  return __builtin_amdgcn_wmma_f32_16x16x4_f32(
      /*neg_a=*/false, a, /*neg_b=*/false, b,
      /*c_mod=*/(short)0, c, /*reuse_a=*/false, /*reuse_b=*/false);
}

// ---------------------------------------------------------------------------
// Edge block: MLP(19->16 relu ->2) per edge, fused with scatter-add to agg[row]
// and block-partial sums for the edge mean. One wave = 16 edges.
// ---------------------------------------------------------------------------
__global__ void __launch_bounds__(256)
gn_edge_kernel(const float* __restrict__ x, const int* __restrict__ eidx,
               const float* __restrict__ ein, const float* __restrict__ gptr,
               const float* __restrict__ W1, const float* __restrict__ b1,
               const float* __restrict__ W2, const float* __restrict__ b2,
               float* __restrict__ eout, float* __restrict__ agg,
               float* __restrict__ esum) {
  __shared__ float smem[8 * 320];   // per-wave: 16x20 features, reused as 16x17 hidden
  __shared__ float bsum[GN_EDGE_F];
  const int tid  = threadIdx.x;
  const int lane = tid & 31;
  const int w    = tid >> 5;
  const int m    = lane & 15;
  const bool hi  = lane >= 16;
  float* feat = &smem[w * 320];

  if (tid < GN_EDGE_F) bsum[tid] = 0.0f;

  const int e = (blockIdx.x * 8 + w) * 16 + m;   // grid covers E exactly
  int row = 0;
  // stage [g | x[row] | x[col] | eattr | 0pad] into the wave's 16x20 LDS tile
  if (!hi) {
    row = eidx[e];
    feat[m * 20 + 0] = gptr[0];
    const float* xr = &x[row * GN_NODE_F];
#pragma unroll
    for (int j = 0; j < GN_NODE_F; ++j) feat[m * 20 + 1 + j] = xr[j];
  } else {
    const int col = eidx[GN_N_EDGES + e];
    const float* xc = &x[col * GN_NODE_F];
#pragma unroll
    for (int j = 0; j < GN_NODE_F; ++j) feat[m * 20 + 9 + j] = xc[j];
    feat[m * 20 + 17] = ein[e * 2 + 0];
    feat[m * 20 + 18] = ein[e * 2 + 1];
    feat[m * 20 + 19] = 0.0f;
  }
  __syncthreads();

  // layer 1: [16x20] @ [20x16] as 5 x V_WMMA_F32_16X16X4_F32
  const int n    = m;              // this lane's output column
  const int ksel = hi ? 2 : 0;     // A 16x4 layout: lanes<16 -> K0,K1 ; lanes>=16 -> K2,K3
  v2f afrag[5];
#pragma unroll
  for (int c = 0; c < 5; ++c) {
    const int k0 = 4 * c + ksel;
    afrag[c].x = feat[m * 20 + k0];
    afrag[c].y = feat[m * 20 + k0 + 1];
  }
  v8f acc = {};
#pragma unroll
  for (int c = 0; c < 5; ++c) {
    const int k0 = 4 * c + ksel;
    v2f bfrag;
    bfrag.x = (k0     < GN_EDGE_IN) ? W1[k0 * GN_HID + n]       : 0.0f;
    bfrag.y = (k0 + 1 < GN_EDGE_IN) ? W1[(k0 + 1) * GN_HID + n] : 0.0f;
    acc = gn_wmma_k4(afrag[c], bfrag, acc);
  }
  const float bias = b1[n];
  __syncthreads();
  // bias + ReLU; D layout: VGPR i holds row (i + 8*(lane>=16)), col = lane&15
#pragma unroll
  for (int i = 0; i < 8; ++i) {
    const int mr = i + (hi ? 8 : 0);
    const float h = acc[i] + bias;
    feat[mr * 17 + n] = h > 0.0f ? h : 0.0f;
  }
  __syncthreads();

  // layer 2 (16->2) + store + scatter-add + mean partials, one edge per lane
  if (!hi) {
    float o0 = b2[0], o1 = b2[1];
#pragma unroll
    for (int k = 0; k < GN_HID; ++k) {
      const float h = feat[m * 17 + k];
      o0 += h * W2[k * GN_EDGE_F + 0];
      o1 += h * W2[k * GN_EDGE_F + 1];
    }
    eout[e * 2 + 0] = o0;
    eout[e * 2 + 1] = o1;
    atomicAdd(&agg[row * GN_EDGE_F + 0], o0);
    atomicAdd(&agg[row * GN_EDGE_F + 1], o1);
    atomicAdd(&bsum[0], o0);
    atomicAdd(&bsum[1], o1);
  }
  __syncthreads();
  if (tid < GN_EDGE_F) atomicAdd(&esum[tid], bsum[tid]);
}

// ---------------------------------------------------------------------------
// Node block: MLP(11->16 relu ->8) per node + mean partials. One wave = 16 nodes.
// ---------------------------------------------------------------------------
__global__ void __launch_bounds__(256)
gn_node_kernel(const float* __restrict__ xin, const float* __restrict__ agg,
               const float* __restrict__ gptr,
               const float* __restrict__ W1, const float* __restrict__ b1,
               const float* __restrict__ W2, const float* __restrict__ b2,
               float* __restrict__ xout, float* __restrict__ nsum) {
  __shared__ float smem[8 * 320];
  __shared__ float bsum[GN_NODE_F];
  const int tid  = threadIdx.x;
  const int lane = tid & 31;
  const int w    = tid >> 5;
  const int m    = lane & 15;
  const bool hi  = lane >= 16;
  float* feat = &smem[w * 320];

  if (tid < GN_NODE_F) bsum[tid] = 0.0f;

  int node = (blockIdx.x * 8 + w) * 16 + m;
  const bool valid = node < GN_N_NODES;
  if (!valid) node = GN_N_NODES - 1;   // clamp for safe loads; writes masked below

  if (!hi) {
    feat[m * 12 + 0] = gptr[0];
    const float* xr = &xin[node * GN_NODE_F];
#pragma unroll
    for (int j = 0; j < GN_NODE_F; ++j) feat[m * 12 + 1 + j] = xr[j];
  } else {
    feat[m * 12 + 9]  = agg[node * GN_EDGE_F + 0];
    feat[m * 12 + 10] = agg[node * GN_EDGE_F + 1];
    feat[m * 12 + 11] = 0.0f;
  }
  __syncthreads();

  const int n    = m;
  const int ksel = hi ? 2 : 0;
  v2f afrag[3];
#pragma unroll
  for (int c = 0; c < 3; ++c) {
    const int k0 = 4 * c + ksel;
    afrag[c].x = feat[m * 12 + k0];
    afrag[c].y = feat[m * 12 + k0 + 1];
  }
  v8f acc = {};
#pragma unroll
  for (int c = 0; c < 3; ++c) {
    const int k0 = 4 * c + ksel;
    v2f bfrag;
    bfrag.x = (k0     < GN_NODE_IN) ? W1[k0 * GN_HID + n]       : 0.0f;
    bfrag.y = (k0 + 1 < GN_NODE_IN) ? W1[(k0 + 1) * GN_HID + n] : 0.0f;
    acc = gn_wmma_k4(afrag[c], bfrag, acc);
  }
  const float bias = b1[n];
  __syncthreads();
#pragma unroll
  for (int i = 0; i < 8; ++i) {
    const int mr = i + (hi ? 8 : 0);
    const float h = acc[i] + bias;
    feat[mr * 17 + n] = h > 0.0f ? h : 0.0f;
  }
  __syncthreads();

  if (!hi && valid) {
    float out[GN_NODE_F];
#pragma unroll
    for (int j = 0; j < GN_NODE_F; ++j) out[j] = b2[j];
#pragma unroll
    for (int k = 0; k < GN_HID; ++k) {
      const float h = feat[m * 17 + k];
#pragma unroll
      for (int j = 0; j < GN_NODE_F; ++j) out[j] += h * W2[k * GN_NODE_F + j];
    }
#pragma unroll
    for (int j = 0; j < GN_NODE_F; ++j) {
      xout[node * GN_NODE_F + j] = out[j];
      atomicAdd(&bsum[j], out[j]);
    }
  }
  __syncthreads();
  if (tid < GN_NODE_F) atomicAdd(&nsum[tid], bsum[tid]);
}

// ---------------------------------------------------------------------------
// Global block: 11->16 relu ->1 on the means. ~200 FLOPs; one lane.
// ---------------------------------------------------------------------------
__global__ void gn_glob_kernel(const float* __restrict__ esum,
                               const float* __restrict__ nsum,
                               const float* __restrict__ gin_ptr,
                               const float* __restrict__ W1,
                               const float* __restrict__ b1,
                               const float* __restrict__ W2,
                               const float* __restrict__ b2,
                               float* __restrict__ gout) {
  if (threadIdx.x != 0) return;
  float gin[11];
#pragma unroll
  for (int j = 0; j < 8; ++j) gin[j] = nsum[j] * (1.0f / GN_N_NODES);
  gin[8]  = esum[0] * (1.0f / GN_N_EDGES);
  gin[9]  = esum[1] * (1.0f / GN_N_EDGES);
  gin[10] = gin_ptr[0];          // read old g before (possibly aliased) write
  float o = b2[0];
  for (int k = 0; k < GN_HID; ++k) {
    float h = b1[k];
    for (int j = 0; j < 11; ++j) h += gin[j] * W1[j * GN_HID + k];
    h = h > 0.0f ? h : 0.0f;
    o += h * W2[k];
  }
  gout[0] = o;
}

__global__ void gn_zero_kernel(float* __restrict__ p, int n) {
  const int i = blockIdx.x * blockDim.x + threadIdx.x;
  if (i < n) p[i] = 0.0f;
}

// ---------------------------------------------------------------------------
extern "C" void kernel_launch(void* const* d_in, const int* in_sizes, int n_in,
                              void* d_out, int out_size, void* d_ws, size_t ws_size,
                              hipStream_t stream) {
  const float* x0   = (const float*)d_in[0];
  const int*   eidx = (const int*)d_in[1];
  const float* ea0  = (const float*)d_in[2];
  const float* g0   = (const float*)d_in[3];
  const float* eW1  = (const float*)d_in[4];
  const float* eb1  = (const float*)d_in[5];
  const float* eW2  = (const float*)d_in[6];
  const float* eb2  = (const float*)d_in[7];
  const float* nW1  = (const float*)d_in[8];
  const float* nb1  = (const float*)d_in[9];
  const float* nW2  = (const float*)d_in[10];
  const float* nb2  = (const float*)d_in[11];
  const float* gW1  = (const float*)d_in[12];
  const float* gb1  = (const float*)d_in[13];
  const float* gW2  = (const float*)d_in[14];
  const float* gb2  = (const float*)d_in[15];

  float* ws    = (float*)d_ws;          // ~16.8 MB of float scratch
  float* agg   = ws;                    // N*2        @ 0
  float* esum  = ws + 200000;           // 2
  float* nsum  = ws + 200002;           // 8
  float* ws_x  = ws + 200016;           // N*8
  float* ws_e  = ws + 1000016;          // E*2
  float* g_ws  = ws + 4200016;          // 1

  for (int l = 0; l < 3; ++l) {
    // zero agg + esum + nsum (contiguous 200010 floats)
    gn_zero_kernel<<<(200010 + 255) / 256, 256, 0, stream>>>(agg, 200010);

    const float* gp  = (l == 0) ? g0  : g_ws;
    const float* xin = (l == 0) ? x0  : ws_x;
    const float* ein = (l == 0) ? ea0 : ws_e;

    gn_edge_kernel<<<GN_N_EDGES / 128, 256, 0, stream>>>(
        xin, eidx, ein, gp,
        eW1 + l * GN_EDGE_IN * GN_HID, eb1 + l * GN_HID,
        eW2 + l * GN_HID * GN_EDGE_F,  eb2 + l * GN_EDGE_F,
        ws_e, agg, esum);

    float* xout = (l == 2) ? (float*)d_out : ws_x;
    gn_node_kernel<<<(GN_N_NODES + 127) / 128, 256, 0, stream>>>(
        xin, agg, gp,
        nW1 + l * GN_NODE_IN * GN_HID, nb1 + l * GN_HID,
        nW2 + l * GN_HID * GN_NODE_F,  nb2 + l * GN_NODE_F,
        xout, nsum);

    gn_glob_kernel<<<1, 32, 0, stream>>>(
        esum, nsum, gp,
        gW1 + l * 11 * GN_HID, gb1 + l * GN_HID,
        gW2 + l * GN_HID,      gb2 + l, g_ws);
  }
}